// mLSTMCellCore_197568495841
// MI455X (gfx1250) — compile-verified
//
#include <hip/hip_runtime.h>
#include <math.h>

typedef float v2f __attribute__((ext_vector_type(2)));
typedef float v8f __attribute__((ext_vector_type(8)));
typedef int   v4i __attribute__((vector_size(16)));   // matches builtin param

#define GPTR __attribute__((address_space(1)))
#define LPTR __attribute__((address_space(3)))

#define A_GATE      15.0f
#define DENOM_FLOOR 1.0f
#define LN_EPS      1e-5f

constexpr int B_   = 128;
constexpr int H_   = 8;
constexpr int D_   = 256;   // dqk == dhv

// Async-LDS staging support (gfx1250): probe the toolchain.
#if defined(__has_builtin)
#  if __has_builtin(__builtin_amdgcn_global_load_async_to_lds_b128)
#    define HAVE_ASYNC_LDS 1
#  else
#    define HAVE_ASYNC_LDS 0
#  endif
#  if __has_builtin(__builtin_amdgcn_s_wait_asynccnt)
#    define HAVE_WAIT_ASYNC_BUILTIN 1
#  else
#    define HAVE_WAIT_ASYNC_BUILTIN 0
#  endif
#else
#  define HAVE_ASYNC_LDS 0
#  define HAVE_WAIT_ASYNC_BUILTIN 0
#endif

__device__ __forceinline__ void wait_async_zero() {
#if HAVE_ASYNC_LDS
#  if HAVE_WAIT_ASYNC_BUILTIN
    __builtin_amdgcn_s_wait_asynccnt(0);
#  else
    asm volatile("s_wait_asynccnt 0" ::: "memory");
#  endif
#endif
}

constexpr int SROWS  = 16;              // rows of C staged per stage
constexpr int NSTAGE = D_ / SROWS;      // 16 stages
constexpr int SLAB   = SROWS * D_;      // 4096 floats = 16KB per buffer

// One workgroup (256 threads = 8 waves) per (b,h) pair.
//  - C is streamed once: global -> (async LDS staging, b128) -> gated FMA ->
//    nontemporal b32 stores of C_new, with the readout y = C_new^T * q_s
//    accumulated in-flight via chained V_WMMA_F32_16X16X4_F32.
__global__ __launch_bounds__(256)
void mlstm_cell_fused(const float* __restrict__ q,
                      const float* __restrict__ k,
                      const float* __restrict__ v,
                      const float* __restrict__ o,
                      const float* __restrict__ i_pre,
                      const float* __restrict__ f_pre,
                      const float* __restrict__ Cin,
                      const float* __restrict__ n,
                      const float* __restrict__ m,
                      const float* __restrict__ gamma,
                      const float* __restrict__ beta,
                      float* __restrict__ out_h,
                      float* __restrict__ out_C,
                      float* __restrict__ out_n,
                      float* __restrict__ out_m)
{
#if HAVE_ASYNC_LDS
    __shared__ float s_C[2][SLAB];      // 32KB double buffer for C staging
#endif
    __shared__ float s_k[D_];
    __shared__ float s_q[D_];           // q / sqrt(dqk)
    __shared__ float s_y[D_];
    __shared__ float s_is, s_fs, s_denm;
    __shared__ float s_den, s_sum, s_ssq;

    const int bh   = blockIdx.x;        // 0..1023
    const int tid  = threadIdx.x;       // 0..255
    const int base = bh * D_;
    const float* __restrict__ Cb  = Cin   + (size_t)bh * D_ * D_;
    float*       __restrict__ Cob = out_C + (size_t)bh * D_ * D_;

    // ---- scalar gate math (stabilized exponential gating) ----
    if (tid == 0) {
        float ic    = tanhf(i_pre[bh] * (1.0f / A_GATE)) * A_GATE;   // softcap
        float fc    = tanhf(f_pre[bh] * (1.0f / A_GATE)) * A_GATE;
        float log_i = ic;
        float log_f = -log1pf(expf(-fc));        // -softplus(-fc)
        float m_old = m[bh];
        float mt    = fmaxf(log_f + m_old, log_i);
        s_is   = expf(log_i - mt);
        s_fs   = expf(log_f + m_old - mt);
        s_denm = expf(-mt);
        s_den  = 0.0f; s_sum = 0.0f; s_ssq = 0.0f;
        out_m[bh] = mt;
    }
    s_k[tid] = k[base + tid];
    s_q[tid] = q[base + tid] * 0.0625f;          // 1/sqrt(256)

#if HAVE_ASYNC_LDS
    // Kick off stage 0 of the C stream (b128 per lane, ASYNCcnt-tracked).
    {
        #pragma unroll
        for (int i = 0; i < 4; ++i) {
            const int f = i * 1024 + tid * 4;    // flat float index in slab
            __builtin_amdgcn_global_load_async_to_lds_b128(
                (GPTR v4i*)(Cb + f),
                (LPTR v4i*)(&s_C[0][f]),
                0, 0);
        }
    }
    wait_async_zero();
#endif
    __syncthreads();

    const float i_s = s_is;
    const float f_s = s_fs;

    // ---- n_new and denominator partial (one element per thread) ----
    {
        float nn = fmaf(f_s, n[base + tid], i_s * s_k[tid]);
        out_n[base + tid] = nn;
        atomicAdd(&s_den, nn * s_q[tid]);        // ds_add_f32
    }

    // ---- fused C stream + WMMA readout ----
    const int wave = tid >> 5;
    const int lane = tid & 31;
    const int half = lane >> 4;                  // K-group within chunk
    const int lm   = lane & 15;                  // M index (= e within tile)
    const int e0   = wave * 32;

    const float vA = v[base + e0 + lm];          // tile-0 column value of v
    const float vB = v[base + e0 + 16 + lm];     // tile-1 column value of v

    v8f acc0 = {};
    v8f acc1 = {};

#if HAVE_ASYNC_LDS
    int buf = 0;
    for (int s = 0; s < NSTAGE; ++s) {
        // Prefetch next stage into the other buffer while consuming this one.
        if (s + 1 < NSTAGE) {
            const float* gsrc = Cb + (size_t)(s + 1) * SLAB;
            #pragma unroll
            for (int i = 0; i < 4; ++i) {
                const int f = i * 1024 + tid * 4;
                __builtin_amdgcn_global_load_async_to_lds_b128(
                    (GPTR v4i*)(gsrc + f),
                    (LPTR v4i*)(&s_C[buf ^ 1][f]),
                    0, 0);
            }
        }

        const int rowbase = s * SROWS;
        #pragma unroll
        for (int dd = 0; dd < SROWS; dd += 4) {
            const int r0  = rowbase + dd + 2 * half; // global K rows of chunk
            const int lr0 = dd + 2 * half;           // local rows in slab
            const float qk0 = s_q[r0 + 0];
            const float qk1 = s_q[r0 + 1];
            const float t0  = i_s * s_k[r0 + 0];
            const float t1  = i_s * s_k[r0 + 1];

            // tile 0: columns e0 .. e0+15
            {
                const int col = e0 + lm;
                float c0 = s_C[buf][(lr0 + 0) * D_ + col];
                float c1 = s_C[buf][(lr0 + 1) * D_ + col];
                float n0 = fmaf(f_s, c0, t0 * vA);
                float n1 = fmaf(f_s, c1, t1 * vA);
                __builtin_nontemporal_store(n0, &Cob[(size_t)(r0 + 0) * D_ + col]);
                __builtin_nontemporal_store(n1, &Cob[(size_t)(r0 + 1) * D_ + col]);
                v2f a; a.x = n0;  a.y = n1;     // A[m][k] = C_new[d0+k][e0+m]
                v2f b; b.x = qk0; b.y = qk1;    // B[k][n] = q_s[d0+k]
                acc0 = __builtin_amdgcn_wmma_f32_16x16x4_f32(
                    false, a, false, b, (short)0, acc0, false, false);
            }
            // tile 1: columns e0+16 .. e0+31
            {
                const int col = e0 + 16 + lm;
                float c0 = s_C[buf][(lr0 + 0) * D_ + col];
                float c1 = s_C[buf][(lr0 + 1) * D_ + col];
                float n0 = fmaf(f_s, c0, t0 * vB);
                float n1 = fmaf(f_s, c1, t1 * vB);
                __builtin_nontemporal_store(n0, &Cob[(size_t)(r0 + 0) * D_ + col]);
                __builtin_nontemporal_store(n1, &Cob[(size_t)(r0 + 1) * D_ + col]);
                v2f a; a.x = n0;  a.y = n1;
                v2f b; b.x = qk0; b.y = qk1;
                acc1 = __builtin_amdgcn_wmma_f32_16x16x4_f32(
                    false, a, false, b, (short)0, acc1, false, false);
            }
        }

        wait_async_zero();      // next buffer's data landed (this wave's issues)
        __syncthreads();        // all waves done reading buf, all data visible
        buf ^= 1;
    }
#else
    // Fallback: direct (nontemporal) global streaming.
    for (int d0 = 0; d0 < D_; d0 += 4) {
        const int r0 = d0 + 2 * half;
        const float qk0 = s_q[r0 + 0];
        const float qk1 = s_q[r0 + 1];
        const float t0  = i_s * s_k[r0 + 0];
        const float t1  = i_s * s_k[r0 + 1];
        {
            const int col = e0 + lm;
            float c0 = __builtin_nontemporal_load(&Cb[(size_t)(r0 + 0) * D_ + col]);
            float c1 = __builtin_nontemporal_load(&Cb[(size_t)(r0 + 1) * D_ + col]);
            float n0 = fmaf(f_s, c0, t0 * vA);
            float n1 = fmaf(f_s, c1, t1 * vA);
            __builtin_nontemporal_store(n0, &Cob[(size_t)(r0 + 0) * D_ + col]);
            __builtin_nontemporal_store(n1, &Cob[(size_t)(r0 + 1) * D_ + col]);
            v2f a; a.x = n0;  a.y = n1;
            v2f b; b.x = qk0; b.y = qk1;
            acc0 = __builtin_amdgcn_wmma_f32_16x16x4_f32(
                false, a, false, b, (short)0, acc0, false, false);
        }
        {
            const int col = e0 + 16 + lm;
            float c0 = __builtin_nontemporal_load(&Cb[(size_t)(r0 + 0) * D_ + col]);
            float c1 = __builtin_nontemporal_load(&Cb[(size_t)(r0 + 1) * D_ + col]);
            float n0 = fmaf(f_s, c0, t0 * vB);
            float n1 = fmaf(f_s, c1, t1 * vB);
            __builtin_nontemporal_store(n0, &Cob[(size_t)(r0 + 0) * D_ + col]);
            __builtin_nontemporal_store(n1, &Cob[(size_t)(r0 + 1) * D_ + col]);
            v2f a; a.x = n0;  a.y = n1;
            v2f b; b.x = qk0; b.y = qk1;
            acc1 = __builtin_amdgcn_wmma_f32_16x16x4_f32(
                false, a, false, b, (short)0, acc1, false, false);
        }
    }
#endif

    // D layout: lane(0-15)=N, VGPR r -> M=r (lanes 0-15) / M=r+8 (lanes 16-31).
    // All N columns identical (B column-broadcast): lane lm==0 of each half
    // extracts y[e0 + half*8 + r].
    if (lm == 0) {
        #pragma unroll
        for (int r = 0; r < 8; ++r) {
            s_y[e0 +      half * 8 + r] = acc0[r];
            s_y[e0 + 16 + half * 8 + r] = acc1[r];
        }
    }
    __syncthreads();

    // ---- LayerNorm statistics over dhv ----
    const float yv = s_y[tid];
    atomicAdd(&s_sum, yv);
    atomicAdd(&s_ssq, yv * yv);
    __syncthreads();

    const float mu    = s_sum * (1.0f / D_);
    const float var   = s_ssq * (1.0f / D_) - mu * mu;
    const float y_ln  = (yv - mu) * rsqrtf(var + LN_EPS) * gamma[tid] + beta[tid];
    const float denom = fmaxf(fmaxf(fabsf(s_den), s_denm), DENOM_FLOOR);
    out_h[base + tid] = o[base + tid] * (y_ln / denom);
}

extern "C" void kernel_launch(void* const* d_in, const int* in_sizes, int n_in,
                              void* d_out, int out_size, void* d_ws, size_t ws_size,
                              hipStream_t stream) {
    (void)in_sizes; (void)n_in; (void)out_size; (void)d_ws; (void)ws_size;

    const float* q     = (const float*)d_in[0];
    const float* k     = (const float*)d_in[1];
    const float* v     = (const float*)d_in[2];
    const float* o     = (const float*)d_in[3];
    const float* i_pre = (const float*)d_in[4];
    const float* f_pre = (const float*)d_in[5];
    const float* C     = (const float*)d_in[6];
    const float* n     = (const float*)d_in[7];
    const float* m     = (const float*)d_in[8];
    const float* gamma = (const float*)d_in[9];
    const float* beta  = (const float*)d_in[10];

    // Output tuple concatenated flat: h, C_new, n_new, m_t
    const size_t h_elems = (size_t)B_ * H_ * D_;        // 262144
    const size_t C_elems = (size_t)B_ * H_ * D_ * D_;   // 67108864
    const size_t n_elems = (size_t)B_ * H_ * D_;        // 262144

    float* out_h = (float*)d_out;
    float* out_C = out_h + h_elems;
    float* out_n = out_C + C_elems;
    float* out_m = out_n + n_elems;

    dim3 grid(B_ * H_);     // 1024 workgroups, one per (b,h)
    dim3 block(256);        // 8 waves
    mlstm_cell_fused<<<grid, block, 0, stream>>>(
        q, k, v, o, i_pre, f_pre, C, n, m, gamma, beta,
        out_h, out_C, out_n, out_m);
}